// TestNet_78606491452206
// MI455X (gfx1250) — compile-verified
//
#include <hip/hip_runtime.h>

// snntorch Synaptic neuron scan, reset_mechanism='zero'
//   reset  = (mem_prev > thr)
//   syn'   = alpha*syn + x_t
//   mem'   = beta*mem + syn'
//   syn'  *= (1-reset); mem' *= (1-reset)
//   spk    = (mem' > thr)
// Outputs concatenated flat: spk[T*M], syn[T*M], mem[T*M]

#define ALPHA 0.1f
#define BETA  0.96f
#define V_THR 2.0f

constexpr int T     = 512;
constexpr int Bsz   = 32;
constexpr int N     = 4096;
constexpr int M     = Bsz * N;      // 131072 independent scan lanes
constexpr int VEC   = 4;            // floats per thread (v4f / b128)
constexpr int BLK   = 64;           // threads per block (2 waves)
constexpr int TILE  = BLK * VEC;    // 256 floats = 1 KB staged per timestep
constexpr int DEPTH = 16;           // async pipeline depth (LDS ring buffers)

// Native clang vector types (HIP float4 is a class and unusable with
// __builtin_nontemporal_store / the async-builtin pointee types).
typedef float v4f __attribute__((ext_vector_type(4)));
typedef int   v4i __attribute__((vector_size(16)));
typedef __attribute__((address_space(1))) v4i* g_b128_ptr;   // global
typedef __attribute__((address_space(3))) v4i* l_b128_ptr;   // LDS

// ---- CDNA5 async copy: global -> LDS, 16 bytes per lane, ASYNCcnt-tracked ----
// cpol = 1 -> TH_LOAD_NT: x is streamed exactly once, don't cache it.
__device__ __forceinline__ void async_tile_b128(const float* gsrc, float* ldst) {
#if __has_builtin(__builtin_amdgcn_global_load_async_to_lds_b128)
  __builtin_amdgcn_global_load_async_to_lds_b128(
      (g_b128_ptr)(void*)(gsrc),
      (l_b128_ptr)(void*)(ldst),
      /*offset=*/0, /*cpol=*/1 /* TH_LOAD_NT */);
#else
  unsigned loff = (unsigned)(unsigned long long)
      (__attribute__((address_space(3))) char*)(ldst);
  asm volatile("global_load_async_to_lds_b128 %0, %1, off th:TH_LOAD_NT"
               :: "v"(loff), "v"((const void*)gsrc)
               : "memory");
#endif
}

// In-order async completion => waiting ASYNCcnt <= n guarantees all but the
// newest n issued async ops have landed in LDS.  "memory" clobber pins the
// subsequent ds_load after the wait and the async issue before it.
#define WAIT_ASYNC(n) asm volatile("s_wait_asynccnt %0" :: "n"(n) : "memory")
// Close the cross-pipe WAR window: the wave's ds_load of ring slot s (iter t)
// must retire before the async engine may overwrite slot s (prefetch issued
// at iter t+1).  DScnt is ~always 0 here already, so this is free.
#define WAIT_DS0() asm volatile("s_wait_dscnt 0x0" ::: "memory")

__global__ __launch_bounds__(BLK) void snn_synaptic_scan_kernel(
    const float* __restrict__ x, float* __restrict__ out) {
  __shared__ float lds[DEPTH * TILE];   // 16 KB ring of staged x tiles

  const int tid   = threadIdx.x;
  const int mBase = blockIdx.x * TILE + tid * VEC;   // this thread's 4 lanes

  const float* __restrict__ xg    = x + mBase;                 // + t*M per step
  float*       __restrict__ spk_o = out + mBase;
  float*       __restrict__ syn_o = out + (size_t)T * M + mBase;
  float*       __restrict__ mem_o = out + 2ull * (size_t)T * M + mBase;
  float* lslot = &lds[tid * VEC];       // this thread's slot in each ring buffer

  // Prologue: fill the pipeline with tiles 0 .. DEPTH-2 (DEPTH-1 in flight).
#pragma unroll
  for (int d = 0; d < DEPTH - 1; ++d)
    async_tile_b128(xg + (size_t)d * M, lslot + d * TILE);

  v4f syn = {0.f, 0.f, 0.f, 0.f};
  v4f mem = {0.f, 0.f, 0.f, 0.f};

  for (int t = 0; t < T; ++t) {
    const int pf = t + DEPTH - 1;
    if (pf < T) {
      WAIT_DS0();              // ds read of this ring slot (iter pf-DEPTH) retired
      async_tile_b128(xg + (size_t)pf * M, lslot + (pf & (DEPTH - 1)) * TILE);
      WAIT_ASYNC(DEPTH - 1);   // in-order completion => tiles 0..t resident
    } else {
      WAIT_ASYNC(0);           // drain tail of the pipeline
    }

    const v4f xv = *(const v4f*)(lslot + (t & (DEPTH - 1)) * TILE);

    v4f spk;
#define SNN_STEP(c)                                              \
    {                                                            \
      const float keep = (mem.c > V_THR) ? 0.0f : 1.0f;          \
      float s = ALPHA * syn.c + xv.c;                            \
      float m = BETA  * mem.c + s;                               \
      s *= keep;                                                 \
      m *= keep;                                                 \
      syn.c = s;                                                 \
      mem.c = m;                                                 \
      spk.c = (m > V_THR) ? 1.0f : 0.0f;                         \
    }
    SNN_STEP(x) SNN_STEP(y) SNN_STEP(z) SNN_STEP(w)
#undef SNN_STEP

    // Streaming outputs: write-once, never re-read -> non-temporal b128 stores
    const size_t o = (size_t)t * M;
    __builtin_nontemporal_store(spk, (v4f*)(spk_o + o));
    __builtin_nontemporal_store(syn, (v4f*)(syn_o + o));
    __builtin_nontemporal_store(mem, (v4f*)(mem_o + o));
  }
}

extern "C" void kernel_launch(void* const* d_in, const int* in_sizes, int n_in,
                              void* d_out, int out_size, void* d_ws, size_t ws_size,
                              hipStream_t stream) {
  (void)in_sizes; (void)n_in; (void)d_ws; (void)ws_size; (void)out_size;
  const float* x = (const float*)d_in[0];
  float* out = (float*)d_out;
  dim3 grid(M / TILE);   // 512 blocks
  dim3 block(BLK);       // 64 threads = 2 waves
  snn_synaptic_scan_kernel<<<grid, block, 0, stream>>>(x, out);
}